// AtomScheduler_15779709845959
// MI455X (gfx1250) — compile-verified
//
#include <hip/hip_runtime.h>

// AtomScheduler forward: out[b,c,t] = (t >= s[b,c]) ? items[b,c,t-s] : 0,
// s = (int)(positions[b,c] * N). Shifted row copy + zero prefix.
// Traffic floor: 128 MiB write + ~(N-s) reads (~64 MiB avg) -> ~8.6us @ 23.3 TB/s.
//
// CDNA5 path: async global->LDS b128 DMA (ASYNCcnt, GVS saddr form) of a
// 16B-aligned input window, s_wait_asynccnt + barrier, LDS drain (sub-16B
// shift absorbed at 4B granularity), aligned global_store_b128.
// All specialization branches are uniform per block (s, t0, m uniform).

#define B_   16
#define C_   64
#define N_   32768
#define CH   4096            // fp32 output elements per workgroup chunk
#define TPB  256             // 8 wave32s
#define CHUNKS (N_ / CH)     // 8 chunks per row
#define VPT  (CH / (TPB * 4))  // float4 groups per thread = 4

__global__ __launch_bounds__(TPB)
void atom_place_kernel(const float* __restrict__ items,
                       const float* __restrict__ positions,
                       float* __restrict__ out)
{
    __shared__ __align__(16) float lds[CH + 8];   // CH+4 used; pad for safety

    const int row = blockIdx.y;            // b*C + c  (uniform)
    const int t0  = blockIdx.x * CH;       // output chunk start within row
    const int tid = threadIdx.x;

    // Truncating cast matches jnp astype(int32) on pos*N (uniform -> SALU).
    const int s = (int)(positions[row] * (float)N_);

    float* rowout = out + (size_t)row * (size_t)N_;

    // ---- Case 1 (uniform): chunk entirely in the zero prefix. No reads. ----
    if (t0 + CH <= s) {
        const float4 z = make_float4(0.0f, 0.0f, 0.0f, 0.0f);
        #pragma unroll
        for (int v = 0; v < VPT; ++v) {
            const int t = t0 + (v * TPB + tid) * 4;
            *(float4*)&rowout[t] = z;
        }
        return;
    }

    const int d  = t0 - s;                 // input index of output element t0
    const int m  = d & 3;                  // sub-16B misalignment (uniform, 0..3)
    const int g0 = d - m;                  // 16B-aligned staging origin

    const unsigned long long rowin =
        (unsigned long long)(const void*)(items + (size_t)row * (size_t)N_);

    // Stage row input from index g0+j (clamped to 0..N-4) into LDS slot j, j in [0, CH+4), 16B per copy.
    // Clamped loads only fill LDS slots that are masked (t < s) or unused tail.
    #pragma unroll
    for (int k = 0; k < VPT; ++k) {
        const int j4 = (k * TPB + tid) * 4;        // 16B-aligned LDS slot
        int gi = g0 + j4;
        gi = gi < 0 ? 0 : gi;
        gi = gi > (N_ - 4) ? (N_ - 4) : gi;        // keeps 16B alignment
        const unsigned voff   = (unsigned)gi * 4u; // byte offset, fits 32b
        const unsigned ldsoff = (unsigned)(size_t)(const void*)&lds[j4];
        asm volatile("global_load_async_to_lds_b128 %0, %1, %2"
                     :: "v"(ldsoff), "v"(voff), "s"(rowin)
                     : "memory");
    }
    // Tail: 4 extra floats at j = CH (covers m in 1..3), single-lane transfer.
    if (tid == 0) {
        int gi = g0 + CH;
        gi = gi < 0 ? 0 : gi;
        gi = gi > (N_ - 4) ? (N_ - 4) : gi;
        const unsigned voff   = (unsigned)gi * 4u;
        const unsigned ldsoff = (unsigned)(size_t)(const void*)&lds[CH];
        asm volatile("global_load_async_to_lds_b128 %0, %1, %2"
                     :: "v"(ldsoff), "v"(voff), "s"(rowin)
                     : "memory");
    }

    asm volatile("s_wait_asynccnt 0" ::: "memory");
    __syncthreads();

    // ---- Case 2 (uniform): chunk entirely past the shift point: no mask. ----
    if (t0 >= s) {
        if (m == 0) {
            #pragma unroll
            for (int v = 0; v < VPT; ++v) {
                const int e4 = (v * TPB + tid) * 4;
                float4 q = *(const float4*)&lds[e4];   // ds_load_b128
                *(float4*)&rowout[t0 + e4] = q;
            }
        } else {
            const float* lp = lds + m;                 // 4B-aligned shifted view
            #pragma unroll
            for (int v = 0; v < VPT; ++v) {
                const int e4 = (v * TPB + tid) * 4;
                float4 q;
                q.x = lp[e4 + 0];
                q.y = lp[e4 + 1];
                q.z = lp[e4 + 2];
                q.w = lp[e4 + 3];
                *(float4*)&rowout[t0 + e4] = q;
            }
        }
        return;
    }

    // ---- Case 3 (uniform, at most one chunk per row): mixed zero/data. ----
    {
        const float* lp = lds + m;                     // m may be 0; 4B path is safe
        #pragma unroll
        for (int v = 0; v < VPT; ++v) {
            const int e4 = (v * TPB + tid) * 4;
            float4 q;
            q.x = lp[e4 + 0];
            q.y = lp[e4 + 1];
            q.z = lp[e4 + 2];
            q.w = lp[e4 + 3];
            const int t = t0 + e4;
            q.x = (t + 0 >= s) ? q.x : 0.0f;
            q.y = (t + 1 >= s) ? q.y : 0.0f;
            q.z = (t + 2 >= s) ? q.z : 0.0f;
            q.w = (t + 3 >= s) ? q.w : 0.0f;
            *(float4*)&rowout[t] = q;
        }
    }
}

extern "C" void kernel_launch(void* const* d_in, const int* in_sizes, int n_in,
                              void* d_out, int out_size, void* d_ws, size_t ws_size,
                              hipStream_t stream) {
    const float* items     = (const float*)d_in[0];   // (B, C, N) fp32
    const float* positions = (const float*)d_in[1];   // (B, C)    fp32
    // d_in[2] = targets: unused in forward pass
    float* outp = (float*)d_out;                      // (B, C, N) fp32

    dim3 grid(CHUNKS, B_ * C_, 1);
    atom_place_kernel<<<grid, TPB, 0, stream>>>(items, positions, outp);
}